// PointPillarsScatter_42777874268601
// MI455X (gfx1250) — compile-verified
//
#include <hip/hip_runtime.h>

// Problem constants (match reference)
#define NXD 512
#define NYD 512
#define CHANS 64
#define PLANE (NXD * NYD)          // 262144 floats per (b,c) plane

typedef __attribute__((ext_vector_type(4))) unsigned int u32x4;
typedef __attribute__((ext_vector_type(8))) int          i32x8;
typedef __attribute__((ext_vector_type(4))) int          i32x4;
typedef __attribute__((ext_vector_type(4))) float        f32x4;
typedef __attribute__((ext_vector_type(2))) float        f32x2;

// ---------------------------------------------------------------------------
// Scatter kernel (placed first so the disasm snippet shows the TDM sequence).
// Each wave32 owns 32 voxels:
//   - TDM (tensor_load_to_lds) DMAs the wave's 32x64 float slab (8KB,
//     contiguous in global) into the wave's private LDS region.
//   - s_wait_tensorcnt 0, then per voxel: lane l reads channels (2l, 2l+1)
//     via one conflict-free ds_load_b64 and issues two NT scattered stores
//     into the (b, c, x, y) canvas (channel planes are 1MB apart).
// Voxel index inside the loop is wave-uniform -> coords loads become s_loads.
// ---------------------------------------------------------------------------
__global__ void __launch_bounds__(256)
scatter_kernel(const float* __restrict__ feat,
               const int*   __restrict__ coords,
               float*       __restrict__ out,
               int n_voxels) {
    __shared__ __align__(16) unsigned char smem[64 * 1024]; // 8 waves * 8KB

    const int lane = threadIdx.x & 31;
    const int wave = __builtin_amdgcn_readfirstlane((int)(threadIdx.x >> 5));
    const int vbase = blockIdx.x * 256 + wave * 32;

    int nv = n_voxels - vbase;
    nv = nv < 0 ? 0 : (nv > 32 ? 32 : nv);
    if (nv == 0) return;

    // LDS byte offset of this wave's staging region. Flat shared pointers
    // carry the LDS offset in the low 32 bits (ISA 10.2 aperture mapping).
    const unsigned lds_off =
        (unsigned)(unsigned long long)(&smem[0]) + (unsigned)wave * 8192u;

    const unsigned long long gaddr =
        (unsigned long long)(const void*)(feat + (size_t)vbase * CHANS);
    const unsigned T = (unsigned)nv * CHANS; // elements, data_size = 4B

    // ---- D# group 0 (128b): count=1, lds_addr, global_addr(57b), type=2 ----
    u32x4 g0;
    g0.x = 1u;                                            // count=1, user mode
    g0.y = lds_off;                                       // lds_addr (bytes)
    g0.z = (unsigned)(gaddr & 0xFFFFFFFFull);             // global_addr[31:0]
    g0.w = ((unsigned)(gaddr >> 32) & 0x1FFFFFFu)         // global_addr[56:32]
           | 0x80000000u;                                 // type = 2 ("image")

    // ---- D# group 1 (256b): 1-D tile of T dwords ----
    i32x8 g1;
    g1[0] = (int)(2u << 16);                 // data_size = 4 bytes; no mask/pad
    g1[1] = (int)((T & 0xFFFFu) << 16);      // tensor_dim0[15:0] @ bit48
    g1[2] = (int)((T >> 16) | (1u << 16));   // tensor_dim0[31:16] | tensor_dim1=1
    g1[3] = (int)(T << 16);                  // tile_dim0 = T
    g1[4] = 0;                               // tile_dim1 = tile_dim2 = 0 (1-D)
    g1[5] = (int)T;                          // tensor_dim0_stride[31:0]
    g1[6] = 0;                               // stride hi / tensor_dim1_stride lo
    g1[7] = 0;

    i32x4 gz4 = {0, 0, 0, 0};                // groups 2/3 unused (<=2D tensor)
    i32x8 gz8 = {0, 0, 0, 0, 0, 0, 0, 0};    // trailing group (6-arg form)

    // TDM issues per-wave regardless of EXEC; each wave waits on its own DMA.
    __builtin_amdgcn_tensor_load_to_lds(g0, g1, gz4, gz4, gz8, 0);
    __builtin_amdgcn_s_wait_tensorcnt(0);

    const size_t wbase = (size_t)wave * 8192u;
    for (int k = 0; k < nv; ++k) {
        const int v = vbase + k;                       // wave-uniform
        const int b = coords[3 * v + 0];
        const int x = coords[3 * v + 1];
        const int y = coords[3 * v + 2];
        const size_t obase = (size_t)b * ((size_t)CHANS * PLANE)
                           + (size_t)x * NYD + (size_t)y;

        // conflict-free ds_load_b64: lane l -> channels 2l, 2l+1
        f32x2 f = *(const f32x2*)&smem[wbase + (size_t)k * 256u
                                             + (size_t)lane * 8u];

        float* p = out + obase + (size_t)(2 * lane) * PLANE;
        __builtin_nontemporal_store(f.x, p);           // channel 2l
        __builtin_nontemporal_store(f.y, p + PLANE);   // channel 2l+1
    }
}

// ---------------------------------------------------------------------------
// Zero the 256MB canvas with non-temporal b128 stores (don't pollute the
// 192MB L2 with zeros; the scatter's random writes come next).
// ---------------------------------------------------------------------------
__global__ void __launch_bounds__(256)
fill_zero_nt(f32x4* __restrict__ out, int n4) {
    int i = blockIdx.x * blockDim.x + threadIdx.x;
    if (i < n4) {
        f32x4 z = {0.f, 0.f, 0.f, 0.f};
        __builtin_nontemporal_store(z, out + i);
    }
}

// ---------------------------------------------------------------------------
extern "C" void kernel_launch(void* const* d_in, const int* in_sizes, int n_in,
                              void* d_out, int out_size, void* d_ws, size_t ws_size,
                              hipStream_t stream) {
    const float* feat   = (const float*)d_in[0];   // [N, 64] fp32
    const int*   coords = (const int*)d_in[1];     // [N, 3] int32 (b, x, y)
    // d_in[2] = batch_size scalar; canvas size is implied by out_size.
    float* out = (float*)d_out;

    const int n_voxels = in_sizes[0] / CHANS;
    const int n4 = out_size / 4;

    fill_zero_nt<<<(n4 + 255) / 256, 256, 0, stream>>>((f32x4*)out, n4);
    scatter_kernel<<<(n_voxels + 255) / 256, 256, 0, stream>>>(
        feat, coords, out, n_voxels);
}